// WeightedEdgeConv_14791867368172
// MI455X (gfx1250) — compile-verified
//
#include <hip/hip_runtime.h>
#include <stdint.h>

// WeightedEdgeConv scatter-sum for MI455X (gfx1250, wave32).
// out[j] += ew[e] * x[i[e]], x/out are [N,64] f32.
// Memory-bound; x (19.2MB) and out (19.2MB) are L2-resident (192MB L2).
// Edge metadata streamed via CDNA5 async global->LDS copies
// (global_load_async_to_lds_b32, ASYNCcnt), double-buffered.

#define TILE 256      // edges staged per block iteration (== blockDim)
#define NFEAT 64      // features per node (2 floats per lane on wave32)

__device__ __forceinline__ void async_ld_b32(const void* gaddr, void* lds) {
  // low 32 bits of a flat pointer into LDS == LDS byte offset (ISA 10.2 aperture map)
  unsigned loff = (unsigned)(uintptr_t)lds;
  asm volatile("global_load_async_to_lds_b32 %0, %1, off"
               :: "v"(loff), "v"((unsigned long long)(uintptr_t)gaddr)
               : "memory");
}

__global__ __launch_bounds__(256) void edge_scatter_kernel(
    const float* __restrict__ x,
    const int*   __restrict__ gsrc,
    const int*   __restrict__ gdst,
    const float* __restrict__ ew,
    float*       __restrict__ out,
    int nEdges) {
  __shared__ int   s_src[2][TILE];
  __shared__ int   s_dst[2][TILE];
  __shared__ float s_w[2][TILE];

  const int tid  = threadIdx.x;
  const int lane = tid & 31;
  const int wv   = tid >> 5;  // wave index in block: 0..7
  const int nTiles = (nEdges + TILE - 1) / TILE;

  int tile = blockIdx.x;
  if (tile >= nTiles) return;

  auto stage = [&](int buf, int t) {
    long long e  = (long long)t * TILE + tid;
    long long ec = (e < nEdges) ? e : (long long)(nEdges - 1);  // clamp: full EXEC, fixed ASYNCcnt
    async_ld_b32(gsrc + ec, &s_src[buf][tid]);
    async_ld_b32(gdst + ec, &s_dst[buf][tid]);
    async_ld_b32(ew   + ec, &s_w[buf][tid]);
  };

  auto edge_body = [&](int buf, int idx) {
    const int   src = s_src[buf][idx];   // LDS broadcast reads (uniform per wave)
    const int   dst = s_dst[buf][idx];
    const float w   = s_w[buf][idx];
    // wave reads one full 256B source row, coalesced float2 per lane
    const float2* xv = (const float2*)(x + (size_t)src * NFEAT);
    const float2  v  = xv[lane];
    float* o = out + (size_t)dst * NFEAT + (lane << 1);
    atomicAdd(o,     v.x * w);           // global_atomic_add_f32 (no-return), resolves in L2
    atomicAdd(o + 1, v.y * w);
  };

  stage(0, tile);
  int buf = 0;
  for (; tile < nTiles; tile += gridDim.x) {
    const int nxt = tile + (int)gridDim.x;
    if (nxt < nTiles) {
      stage(buf ^ 1, nxt);
      // 3 async ops per stage, in-order completion: cnt<=3 => current tile staged
      asm volatile("s_wait_asynccnt 3" ::: "memory");
    } else {
      asm volatile("s_wait_asynccnt 0" ::: "memory");
    }
    __syncthreads();  // all 8 waves' staging visible

    const long long base = (long long)tile * TILE;
    if (base + TILE <= (long long)nEdges) {
      // full tile: no per-edge bounds checks
#pragma unroll 4
      for (int k = 0; k < 32; ++k)
        edge_body(buf, (wv << 5) + k);
    } else {
      // last (partial) tile
      for (int k = 0; k < 32; ++k) {
        const int idx = (wv << 5) + k;
        if (base + idx < (long long)nEdges)
          edge_body(buf, idx);
      }
    }

    __syncthreads();  // buffer reuse safe next iteration
    buf ^= 1;
  }
}

__global__ void zero_out_kernel(float4* __restrict__ out4, int n4) {
  int i = blockIdx.x * blockDim.x + threadIdx.x;
  if (i < n4) out4[i] = make_float4(0.f, 0.f, 0.f, 0.f);
}

extern "C" void kernel_launch(void* const* d_in, const int* in_sizes, int n_in,
                              void* d_out, int out_size, void* d_ws, size_t ws_size,
                              hipStream_t stream) {
  const float* x   = (const float*)d_in[0];
  const int*   g   = (const int*)d_in[1];   // [2, E]: row0 = src i, row1 = dst j
  const float* ew  = (const float*)d_in[2];
  float*       out = (float*)d_out;

  const int nEdges = in_sizes[2];
  const int* gsrc = g;
  const int* gdst = g + nEdges;

  // 1) zero the accumulator (d_out is poisoned by the harness)
  const int n4 = out_size / 4;
  zero_out_kernel<<<(n4 + 255) / 256, 256, 0, stream>>>((float4*)out, n4);

  // 2) scatter-add; grid-stride over edge tiles with double-buffered async staging
  const int nTiles = (nEdges + TILE - 1) / TILE;
  const int grid = (nTiles < 1536) ? nTiles : 1536;
  edge_scatter_kernel<<<grid, 256, 0, stream>>>(x, gsrc, gdst, ew, out, nEdges);
}